// Detector_20452634263579
// MI455X (gfx1250) — compile-verified
//
#include <hip/hip_runtime.h>
#include <stdint.h>

// ---------------------------------------------------------------------------
// YOLOv3 decode for MI455X (gfx1250).
// Memory-bound (116 MB in / 9.5 MB out -> ~5.4us at 23.3 TB/s).
// CDNA5 Tensor Data Mover DMAs the strided NCHW channel tile
// (255 rows x tile-width, row stride = H*W floats) into LDS; each thread
// decodes one anchor from bank-conflict-free LDS reads.
// Grid is 2D (x = spatial tile, y = batch) to avoid integer division.
// ---------------------------------------------------------------------------

#define TILE      64     // spatial positions per workgroup tile
#define NCH       255    // 3 * (5 + 80)
#define NFIELDS   85     // 5 + 80 per anchor
#define NCLASS    80
#define BLOCK_THREADS (3 * TILE)   // one thread per (anchor, pos) = 192 = 6 waves

typedef uint32_t u32;
typedef uint64_t u64;
typedef u32 __attribute__((ext_vector_type(4))) tdm_g0_t;   // D# group 0 (128b)
typedef int  __attribute__((ext_vector_type(8))) tdm_v8i_t; // D# group 1 (256b) / arg5
typedef int  __attribute__((ext_vector_type(4))) tdm_v4i_t; // D# groups 2/3

__global__ __launch_bounds__(BLOCK_THREADS) void yolo_decode_scale(
    const float* __restrict__ in,        // [B, 255, H, W]
    const float* __restrict__ anchors,   // [3, 2]
    const float* __restrict__ thresh_p,  // [1]
    float* __restrict__ out,             // boxes [N,6] then valid [N]
    int W, int HW, float stride_t,
    int rowBase, int totalN)
{
    __shared__ float tile[NCH * TILE];   // 65,280 bytes -> 4 workgroups / WGP (24 waves)

    const int b     = blockIdx.y;
    const int tile0 = blockIdx.x * TILE;
    const int tw    = (HW - tile0 < TILE) ? (HW - tile0) : TILE;  // clamp edge tile

    const int a = threadIdx.x >> 6;         // anchor 0..2
    const int p = threadIdx.x & (TILE - 1); // position within tile

    // Scalar loads issued early so their latency overlaps the TDM transfer.
    const float thresh = *thresh_p;
    const float aw     = anchors[2 * a + 0];
    const float ah     = anchors[2 * a + 1];

    // ---- TDM async tensor load: [tw spatial (contig)] x [255 ch (stride HW)] -> LDS
    if (threadIdx.x == 0) {
        u64 ga  = (u64)(uintptr_t)(in + (u64)b * NCH * HW + (u64)tile0);
        u32 lds = (u32)(uintptr_t)(&tile[0]);   // low 32 bits of generic addr = LDS byte offset

        tdm_g0_t g0;
        g0[0] = 1u;                                           // count = 1 valid descriptor
        g0[1] = lds;                                          // lds_addr (bytes)
        g0[2] = (u32)ga;                                      // global_addr[31:0]
        g0[3] = (u32)((ga >> 32) & 0x1FFFFFFu) | (2u << 30);  // global_addr[56:32] | type=2

        tdm_v8i_t g1;
        g1[0] = (int)(2u << 16);                              // data_size=4B; wg_mask=0; no pad/iter
        g1[1] = (int)(((u32)HW & 0xFFFFu) << 16);             // tensor_dim0[15:0] @ bits 63:48
        g1[2] = (int)((((u32)HW >> 16) & 0xFFFFu)             // tensor_dim0[31:16]
                      | ((u32)NCH << 16));                    // tensor_dim1[15:0] = 255
        g1[3] = (int)(((u32)tw & 0xFFFFu) << 16);             // tensor_dim1 hi=0 | tile_dim0 = tw
        g1[4] = (int)NCH;                                     // tile_dim1 = 255; tile_dim2 = 0
        g1[5] = (int)HW;                                      // tensor_dim0_stride[31:0] (elements)
        g1[6] = 0;                                            // stride0 hi = 0; stride1 lo = 0
        g1[7] = 0;

        tdm_v4i_t gz4 = {0, 0, 0, 0};                         // 2D tensor: groups 2/3 unused
        tdm_v8i_t gz8 = {0, 0, 0, 0, 0, 0, 0, 0};
        __builtin_amdgcn_tensor_load_to_lds(g0, g1, gz4, gz4, gz8, /*cpol=*/0);
    }
    __builtin_amdgcn_s_wait_tensorcnt(0);   // issuing wave waits; others pass trivially
    __syncthreads();

    if (p < tw) {
        // field k of this anchor lives at tile[(a*85 + k)*tw + p]; lane-consecutive -> no bank conflicts
        const float* v = &tile[(a * NFIELDS) * tw + p];

        const float conf = 1.0f / (1.0f + __expf(-v[0]));
        const float sx   = 1.0f / (1.0f + __expf(-v[(size_t)tw]));
        const float sy   = 1.0f / (1.0f + __expf(-v[(size_t)2 * tw]));
        const float ew   = __expf(v[(size_t)3 * tw]);
        const float eh   = __expf(v[(size_t)4 * tw]);

        const int pos = tile0 + p;
        const int gy  = pos / W;
        const int gx  = pos - gy * W;

        const float cx = ((float)gx + sx) * stride_t;
        const float cy = ((float)gy + sy) * stride_t;
        const float bw = aw * ew;
        const float bh = ah * eh;
        const float x1 = cx - bw * 0.5f;
        const float y1 = cy - bh * 0.5f;
        const float x2 = x1 + bw;
        const float y2 = y1 + bh;

        // argmax over 80 classes (first max wins, matching jnp.argmax)
        float best = v[(size_t)5 * tw];
        int   bi   = 0;
        #pragma unroll 8
        for (int k = 1; k < NCLASS; ++k) {
            const float val = v[(size_t)(5 + k) * tw];
            if (val > best) { best = val; bi = k; }
        }

        const int row = rowBase + (b * HW + pos) * 3 + a;
        float* o = out + (size_t)row * 6;
        o[0] = conf;
        o[1] = x1;
        o[2] = y1;
        o[3] = x2;
        o[4] = y2;
        o[5] = (float)bi;
        out[(size_t)totalN * 6 + row] = (conf > thresh) ? 1.0f : 0.0f;
    }
}

extern "C" void kernel_launch(void* const* d_in, const int* in_sizes, int n_in,
                              void* d_out, int out_size, void* d_ws, size_t ws_size,
                              hipStream_t stream) {
    (void)in_sizes; (void)n_in; (void)out_size; (void)d_ws; (void)ws_size;

    const float* o13 = (const float*)d_in[0];
    const float* o26 = (const float*)d_in[1];
    const float* o52 = (const float*)d_in[2];
    const float* a13 = (const float*)d_in[3];
    const float* a26 = (const float*)d_in[4];
    const float* a52 = (const float*)d_in[5];
    const float* th  = (const float*)d_in[6];
    float* out = (float*)d_out;

    const int B = 32;
    const int HW13 = 13 * 13, HW26 = 26 * 26, HW52 = 52 * 52;
    const int N13 = B * HW13 * 3;          // 16,224 rows
    const int N26 = B * HW26 * 3;          // 64,896 rows
    const int N52 = B * HW52 * 3;          // 259,584 rows
    const int totalN = N13 + N26 + N52;    // 340,704 rows

    const int t13 = (HW13 + TILE - 1) / TILE;  // 3
    const int t26 = (HW26 + TILE - 1) / TILE;  // 11
    const int t52 = (HW52 + TILE - 1) / TILE;  // 43

    yolo_decode_scale<<<dim3(t13, B), BLOCK_THREADS, 0, stream>>>(
        o13, a13, th, out, 13, HW13, 32.0f, 0, totalN);
    yolo_decode_scale<<<dim3(t26, B), BLOCK_THREADS, 0, stream>>>(
        o26, a26, th, out, 26, HW26, 16.0f, N13, totalN);
    yolo_decode_scale<<<dim3(t52, B), BLOCK_THREADS, 0, stream>>>(
        o52, a52, th, out, 52, HW52, 8.0f, N13 + N26, totalN);
}